// NeighborhoodAttention_66546223284252
// MI455X (gfx1250) — compile-verified
//
#include <hip/hip_runtime.h>
#include <hip/hip_bf16.h>

typedef float v2f __attribute__((ext_vector_type(2)));
typedef float v8f __attribute__((ext_vector_type(8)));

static __device__ __forceinline__ v8f wmma4(v2f a, v2f b, v8f c) {
  // D = A(16x4 f32) x B(4x16 f32) + C(16x16 f32)
  return __builtin_amdgcn_wmma_f32_16x16x4_f32(false, a, false, b, (short)0, c,
                                               false, false);
}

// ---- CDNA5 async global->LDS staging (ASYNCcnt) with safe fallback --------
#if __has_builtin(__builtin_amdgcn_global_load_async_to_lds_b128) && \
    __has_builtin(__builtin_amdgcn_s_wait_asynccnt)
#define USE_ASYNC_LDS 1
// Parameter type per hipcc diagnostic: pointer to int __vector(4), generic AS.
typedef int async_b128_t __attribute__((vector_size(4 * sizeof(int))));
#endif

static __device__ __forceinline__ void stage16(float* lds, const float* g) {
#if USE_ASYNC_LDS
  __builtin_amdgcn_global_load_async_to_lds_b128(
      (async_b128_t*)g, (async_b128_t*)lds, 0, 0);
#else
  *(float4*)lds = *(const float4*)g;
#endif
}
static __device__ __forceinline__ void async_fence() {
#if USE_ASYNC_LDS
  __builtin_amdgcn_s_wait_asynccnt(0);
#endif
}

// ---------------------------------------------------------------------------
// fp32 WMMA GEMM:  C[M,N] = (A[M,K] @ B[K,N] + bias[N]) * colScale
// Block: 256 threads (8 waves), tile 128x64, waves own 32x32 (2x2 WMMA tiles).
// A staged row-major (async b128); B staged TRANSPOSED so both A and B
// fragments are single contiguous ds_load_b64 reads.
// ---------------------------------------------------------------------------
#define BM 128
#define BN 64
#define BKC 64
#define LDSA_STRIDE 68  // 64 + 4 pad: conflict-free fragment reads
#define LDSB_STRIDE 68

__global__ __launch_bounds__(256)
void gemm_f32_wmma(const float* __restrict__ A, const float* __restrict__ B,
                   const float* __restrict__ bias, float* __restrict__ C,
                   int M, int N, int Kdim, int scale_cols, float scale) {
  __shared__ float sA[BM * LDSA_STRIDE];   // [m][k]
  __shared__ float sBt[BN * LDSB_STRIDE];  // [n][k]  (transposed)

  const int tid = threadIdx.x;
  const int lane = tid & 31;
  const int wave = tid >> 5;
  const int m0 = blockIdx.x * BM;
  const int n0 = blockIdx.y * BN;
  const int wm = (wave & 3) * 32;   // wave's 32-row M sub-tile
  const int wn = (wave >> 2) * 32;  // wave's 32-col N sub-tile

  const int mq = lane & 15;          // fragment row / col within 16
  const int klane = (lane >> 4) * 2; // k-pair selector per WMMA layout

  v8f acc00 = {0, 0, 0, 0, 0, 0, 0, 0};
  v8f acc01 = {0, 0, 0, 0, 0, 0, 0, 0};
  v8f acc10 = {0, 0, 0, 0, 0, 0, 0, 0};
  v8f acc11 = {0, 0, 0, 0, 0, 0, 0, 0};

  const int ldRow = tid >> 2;        // 64 rows per pass, 4 threads per row
  const int ldCol = (tid & 3) * 16;  // 16 floats = 4x float4 per thread/row

  for (int k0 = 0; k0 < Kdim; k0 += BKC) {
    __syncthreads();
    // Stage A[m0 + 0..127][k0 + 0..63] (row-major, async b128)
#pragma unroll
    for (int p = 0; p < 2; ++p) {
      const int row = ldRow + p * 64;
      const float* ga = A + (size_t)(m0 + row) * Kdim + k0 + ldCol;
      float* la = sA + row * LDSA_STRIDE + ldCol;
#pragma unroll
      for (int j = 0; j < 4; ++j) stage16(la + j * 4, ga + j * 4);
    }
    // Stage B[k0 + 0..63][n0 + 0..63] transposed into sBt[n][k]
    {
      const float* gb = B + (size_t)(k0 + ldRow) * N + n0 + ldCol;
#pragma unroll
      for (int j = 0; j < 4; ++j) {
        float4 bv = *(const float4*)(gb + j * 4);
        float* col = sBt + (ldCol + j * 4) * LDSB_STRIDE + ldRow;
        col[0 * LDSB_STRIDE] = bv.x;
        col[1 * LDSB_STRIDE] = bv.y;
        col[2 * LDSB_STRIDE] = bv.z;
        col[3 * LDSB_STRIDE] = bv.w;
      }
    }
    async_fence();
    __syncthreads();

#pragma unroll
    for (int kk = 0; kk < BKC; kk += 4) {
      const int kb = kk + klane;
      v2f af0 = *(const v2f*)(sA + (wm + mq) * LDSA_STRIDE + kb);
      v2f af1 = *(const v2f*)(sA + (wm + 16 + mq) * LDSA_STRIDE + kb);
      v2f bf0 = *(const v2f*)(sBt + (wn + mq) * LDSB_STRIDE + kb);
      v2f bf1 = *(const v2f*)(sBt + (wn + 16 + mq) * LDSB_STRIDE + kb);
      acc00 = wmma4(af0, bf0, acc00);
      acc01 = wmma4(af0, bf1, acc01);
      acc10 = wmma4(af1, bf0, acc10);
      acc11 = wmma4(af1, bf1, acc11);
    }
  }

  // Epilogue: bias + optional column scale, then store 2x2 tiles.
  const int col0 = n0 + wn + mq;
  const int col1 = col0 + 16;
  const float bb0 = bias[col0];
  const float bb1 = bias[col1];
  const float s0 = (col0 < scale_cols) ? scale : 1.0f;
  const float s1 = (col1 < scale_cols) ? scale : 1.0f;
  const int mrow = m0 + wm + 8 * (lane >> 4);
#pragma unroll
  for (int r = 0; r < 8; ++r) {
    float* cp0 = C + (size_t)(mrow + r) * N;
    float* cp1 = C + (size_t)(mrow + 16 + r) * N;
    cp0[col0] = (acc00[r] + bb0) * s0;
    cp0[col1] = (acc01[r] + bb1) * s1;
    cp1[col0] = (acc10[r] + bb0) * s0;
    cp1[col1] = (acc11[r] + bb1) * s1;
  }
}

// ---------------------------------------------------------------------------
// Neighborhood attention, K=7, H=W=56, d=32, 8 heads, with temporal shift.
// One wave32 per (b, head, y, 16-query x-tile). K staged [key][d] (async),
// V staged TRANSPOSED [d][key] so AV B-fragments are contiguous ds_load_b64.
// ---------------------------------------------------------------------------
#define SPATIAL 56
#define NKEYS 224   // 7 * 32 key slots
#define KPAD 36     // per-key padded float stride for sK
#define VSTRIDE 228 // per-d-row padded stride for sVt
#define SSTRIDE 228 // per-query padded score row stride

__global__ __launch_bounds__(32)
void natt2d_wmma(const float* __restrict__ qkv, const float* __restrict__ rpb,
                 float* __restrict__ out) {
  __shared__ float sK[NKEYS * KPAD];    // [key][d]
  __shared__ float sVt[32 * VSTRIDE];   // [d][key] (transposed)
  __shared__ float sS[16 * SSTRIDE];    // [query][key]

  const int lane = threadIdx.x;
  int bid = blockIdx.x;
  const int xt = bid & 3;  bid >>= 2;
  const int y  = bid % SPATIAL; bid /= SPATIAL;
  const int h  = bid & 7;
  const int b  = bid >> 3;

  const int x0 = (xt == 3) ? 40 : xt * 16;

  // temporal shift: q from min(t,14), k/v from min(t+1,15) within group of 16
  const int g = b >> 4, t = b & 15;
  const int b_pre  = (g << 4) + (t < 14 ? t : 14);
  const int b_post = (g << 4) + (t < 15 ? t + 1 : 15);

  int r0 = y - 3;  if (r0 < 0) r0 = 0; if (r0 > 49) r0 = 49;  // key row start
  int c0 = x0 - 3; if (c0 < 0) c0 = 0; if (c0 > 49) c0 = 49;  // key col start

  // ---- stage K ([key][d], async) and V (transposed [d][key]) ----
  for (int s = lane; s < NKEYS; s += 32) {
    const int rr = s >> 5, cc = s & 31;
    const int row = r0 + rr;
    int col = c0 + cc; if (col > 55) col = 55;  // clamped pads get prob 0
    const size_t tok = ((size_t)(b_post * SPATIAL + row)) * SPATIAL + col;
    const float* kp = qkv + tok * 768 + 256 + h * 32;
    const float* vp = kp + 256;
    float* dK = sK + s * KPAD;
#pragma unroll
    for (int d4 = 0; d4 < 8; ++d4) stage16(dK + d4 * 4, kp + d4 * 4);
#pragma unroll
    for (int d4 = 0; d4 < 8; ++d4) {
      float4 vv = *(const float4*)(vp + d4 * 4);
      sVt[(d4 * 4 + 0) * VSTRIDE + s] = vv.x;
      sVt[(d4 * 4 + 1) * VSTRIDE + s] = vv.y;
      sVt[(d4 * 4 + 2) * VSTRIDE + s] = vv.z;
      sVt[(d4 * 4 + 3) * VSTRIDE + s] = vv.w;
    }
  }

  // ---- load Q A-fragments straight from global ----
  const int mq = lane & 15;
  const int klane = (lane >> 4) * 2;
  const int xq = x0 + mq;
  const size_t tokq = ((size_t)(b_pre * SPATIAL + y)) * SPATIAL + xq;
  const float* qp = qkv + tokq * 768 + h * 32;
  v2f afrag[8];
#pragma unroll
  for (int kc = 0; kc < 8; ++kc)
    afrag[kc] = *(const v2f*)(qp + kc * 4 + klane);

  async_fence();
  __syncthreads();

  // ---- scores: S[16 q][224 keys] via 14 tiles x 8 k-steps of WMMA ----
  v8f S[14];
#pragma unroll
  for (int i = 0; i < 14; ++i) S[i] = (v8f){0, 0, 0, 0, 0, 0, 0, 0};
#pragma unroll
  for (int tile = 0; tile < 14; ++tile) {
    const int rr = tile >> 1, jh = tile & 1;
    const float* kb = sK + ((rr * 32 + jh * 16 + mq) * KPAD) + klane;
#pragma unroll
    for (int kc = 0; kc < 8; ++kc) {
      v2f bf = *(const v2f*)(kb + kc * 4);
      S[tile] = wmma4(afrag[kc], bf, S[tile]);
    }
  }

  // ---- spill scores to LDS (also the C-layout -> A-layout transpose) ----
#pragma unroll
  for (int tile = 0; tile < 14; ++tile) {
    const int rr = tile >> 1, jh = tile & 1;
    const int base = rr * 32 + jh * 16 + mq;
#pragma unroll
    for (int r = 0; r < 8; ++r) {
      const int m = r + 8 * (lane >> 4);
      sS[m * SSTRIDE + base] = S[tile][r];
    }
  }
  __syncthreads();

  // ---- rpb + masked softmax, one query row per lane (lanes 0..15) ----
  if (lane < 16) {
    const int m = lane;
    const int xq2 = x0 + m;
    int njq = xq2 - 3; if (njq < 0) njq = 0; if (njq > 49) njq = 49;
    const int j0 = njq - c0;             // valid 7-wide window start in strip
    const int bi0 = 6 + r0 - y;          // rpb row base index
    const int bj0 = 6 + njq - xq2;       // rpb col base index
    const float* rp = rpb + h * 169;
    float* srow = sS + m * SSTRIDE;

    float mx = -1e30f;
    for (int rr = 0; rr < 7; ++rr)
      for (int u = 0; u < 7; ++u) {
        float sv = srow[rr * 32 + j0 + u] + rp[(bi0 + rr) * 13 + bj0 + u];
        mx = fmaxf(mx, sv);
      }
    float sum = 0.0f;
    for (int rr = 0; rr < 7; ++rr)
      for (int u = 0; u < 7; ++u) {
        float sv = srow[rr * 32 + j0 + u] + rp[(bi0 + rr) * 13 + bj0 + u];
        sum += __expf(sv - mx);
      }
    const float inv = 1.0f / sum;
    for (int c = 0; c < NKEYS; ++c) {
      const int rr = c >> 5;
      const int u = (c & 31) - j0;
      float val = 0.0f;
      if (u >= 0 && u < 7) {
        float sv = srow[c] + rp[(bi0 + rr) * 13 + bj0 + u];
        val = __expf(sv - mx) * inv;
      }
      srow[c] = val;  // padded/invalid keys contribute exactly 0 to AV
    }
  }
  __syncthreads();

  // ---- AV: out[16 q][32 d] = P(16x224) @ V(224x32), fully unrolled ----
  v8f O0 = (v8f){0, 0, 0, 0, 0, 0, 0, 0};
  v8f O1 = (v8f){0, 0, 0, 0, 0, 0, 0, 0};
  const float* ap = sS + mq * SSTRIDE + klane;
  const float* vt0 = sVt + mq * VSTRIDE + klane;        // d col = mq
  const float* vt1 = sVt + (16 + mq) * VSTRIDE + klane; // d col = mq+16
#pragma unroll
  for (int kc = 0; kc < 56; ++kc) {
    v2f af = *(const v2f*)(ap + kc * 4);
    v2f bf0 = *(const v2f*)(vt0 + kc * 4);
    v2f bf1 = *(const v2f*)(vt1 + kc * 4);
    O0 = wmma4(af, bf0, O0);
    O1 = wmma4(af, bf1, O1);
  }

  // ---- store: attn output laid out [token][head*32 + d] for proj GEMM ----
  const size_t tok0 = ((size_t)(b * SPATIAL + y)) * SPATIAL + x0;
#pragma unroll
  for (int r = 0; r < 8; ++r) {
    const int m = r + 8 * (lane >> 4);
    float* op = out + (tok0 + m) * 256 + h * 32;
    op[mq] = O0[r];
    op[16 + mq] = O1[r];
  }
}

// ---------------------------------------------------------------------------
extern "C" void kernel_launch(void* const* d_in, const int* in_sizes, int n_in,
                              void* d_out, int out_size, void* d_ws,
                              size_t ws_size, hipStream_t stream) {
  const float* x      = (const float*)d_in[0];
  const float* w_qkv  = (const float*)d_in[1];
  const float* b_qkv  = (const float*)d_in[2];
  const float* rpb    = (const float*)d_in[3];
  const float* w_proj = (const float*)d_in[4];
  const float* b_proj = (const float*)d_in[5];
  float* out = (float*)d_out;

  const int M = 64 * 56 * 56;  // 200704 tokens
  float* qkv  = (float*)d_ws;                       // M x 768
  float* attn = qkv + (size_t)M * 768;              // M x 256

  const float scale = 0.17677669529663687f;  // 32^-0.5

  // 1) QKV projection (q columns pre-scaled by d^-0.5)
  gemm_f32_wmma<<<dim3(M / 128, 768 / 64), dim3(256), 0, stream>>>(
      x, w_qkv, b_qkv, qkv, M, 768, 256, 256, scale);

  // 2) Neighborhood attention with temporal shift + rpb
  natt2d_wmma<<<dim3(64 * 8 * 56 * 4), dim3(32), 0, stream>>>(qkv, rpb, attn);

  // 3) Output projection
  gemm_f32_wmma<<<dim3(M / 128, 256 / 64), dim3(256), 0, stream>>>(
      attn, w_proj, b_proj, out, M, 256, 256, 0, 1.0f);
}